// PhysicsGuidedNuclearModel_25598005084284
// MI455X (gfx1250) — compile-verified
//
#include <hip/hip_runtime.h>
#include <hip/hip_bf16.h>
#include <math.h>

// Problem constants (from reference)
#define BB   32
#define TT   512
#define PP   52
#define DIMC 64
#define EDC  128
#define DSC  16
#define DCC  4
#define BT   (BB * TT)   // 16384 rows in the time path

typedef __attribute__((ext_vector_type(2))) float v2f;
typedef __attribute__((ext_vector_type(8))) float v8f;

__device__ __forceinline__ float gelu_exact(float x) {
    return 0.5f * x * (1.0f + erff(x * 0.70710678118654752440f));
}
__device__ __forceinline__ float sigmoid_(float x) {
    return 1.0f / (1.0f + __expf(-x));
}

// ---------------------------------------------------------------------------
// fp32 WMMA GEMM: C[M,N] = act(A[M,K] @ W^T + bias + resid),  W stored [N,K].
// Assumptions (hold for every call site): M % 16 == 0, K % 4 == 0, all
// leading dims even (so per-lane v2f fragment loads are 8B aligned).
// Each wave computes a 16 x (16*TN) strip: the A fragment (one b64 load) is
// reused across TN WMMAs; TN independent accumulators hide XDL latency.
// NGUARD=true handles ragged N branchlessly (clamped load + cndmask-to-0),
// keeping EXEC all-ones for every V_WMMA as the ISA requires.
// act: 0=none, 1=gelu, 2=sigmoid, 3=BN(eval)+gelu
// ---------------------------------------------------------------------------
template<int TN, bool NGUARD>
__global__ void gemm_wmma_f32(const float* __restrict__ A, int lda,
                              const float* __restrict__ W, int ldw,
                              const float* __restrict__ bias,
                              const float* __restrict__ resid,
                              float* __restrict__ C, int ldc,
                              int M, int N, int K, int act,
                              const float* __restrict__ bn_g,
                              const float* __restrict__ bn_b,
                              const float* __restrict__ bn_m,
                              const float* __restrict__ bn_v)
{
    const int lane   = threadIdx.x & 31;
    const int wave   = threadIdx.x >> 5;
    const int tilesN = (N + 15) >> 4;
    const int groupsN = (tilesN + TN - 1) / TN;
    const int tilesM = M >> 4;
    const int nGroups = tilesM * groupsN;
    const int grp = blockIdx.x * (blockDim.x >> 5) + wave;
    if (grp >= nGroups) return;                 // wave-uniform
    const int tm = grp / groupsN;
    const int gn = grp - tm * groupsN;
    const int row0 = tm << 4;
    const int col0 = gn * (TN * 16);
    const int hlf  = lane >> 4;                 // half selects K pair / M half
    const int l    = lane & 15;

    const float* Ap = A + (size_t)(row0 + l) * lda + (hlf << 1);
    const float* Wp[TN];
    int col[TN];
    v8f acc[TN];
#pragma unroll
    for (int j = 0; j < TN; ++j) {
        col[j] = col0 + j * 16 + l;
        int c = col[j];
        if (NGUARD && c > N - 1) c = N - 1;     // clamp: load stays in-bounds
        Wp[j] = W + (size_t)c * ldw + (hlf << 1);
        v8f z;
#pragma unroll
        for (int e = 0; e < 8; ++e) z[e] = 0.f;
        acc[j] = z;
    }

    for (int k0 = 0; k0 < K; k0 += 4) {
        const v2f a = *(const v2f*)(Ap + k0);   // one global_load_b64
#pragma unroll
        for (int j = 0; j < TN; ++j) {
            v2f b = *(const v2f*)(Wp[j] + k0);  // one global_load_b64
            if (NGUARD) {
                if (col[j] >= N) { b[0] = 0.f; b[1] = 0.f; }  // cndmask, no branch
            }
            acc[j] = __builtin_amdgcn_wmma_f32_16x16x4_f32(false, a, false, b,
                                                           (short)0, acc[j],
                                                           false, false);
        }
    }

#pragma unroll
    for (int j = 0; j < TN; ++j) {
        const int c = col[j];
#pragma unroll
        for (int e = 0; e < 8; ++e) {
            const int r = row0 + e + (hlf << 3);
            if (!NGUARD || c < N) {
                float v = acc[j][e];
                if (bias)  v += bias[c];
                if (resid) v += resid[(size_t)r * ldc + c];
                if (act == 1) {
                    v = gelu_exact(v);
                } else if (act == 2) {
                    v = sigmoid_(v);
                } else if (act == 3) {
                    const float sc = bn_g[c] * rsqrtf(bn_v[c] + 1e-5f);
                    v = (v - bn_m[c]) * sc + bn_b[c];
                    v = gelu_exact(v);
                }
                C[(size_t)r * ldc + c] = v;
            }
        }
    }
}

template<int TN, bool NGUARD>
static void gemm_launch(const float* A, int lda, const float* W, int ldw,
                        const float* bias, const float* resid, float* C, int ldc,
                        int M, int N, int K, int act,
                        const float* g, const float* bb, const float* mm,
                        const float* vv, hipStream_t stream)
{
    const int tilesM = M >> 4;
    const int tilesN = (N + 15) >> 4;
    const int groupsN = (tilesN + TN - 1) / TN;
    const int nGroups = tilesM * groupsN;
    const int wavesPerBlock = 4;
    const int grid = (nGroups + wavesPerBlock - 1) / wavesPerBlock;
    gemm_wmma_f32<TN, NGUARD><<<grid, 32 * wavesPerBlock, 0, stream>>>(
        A, lda, W, ldw, bias, resid, C, ldc, M, N, K, act, g, bb, mm, vv);
}

// out[Cc,R] = in[R,Cc]^T   (tiny; used once per mamba for Bm)
__global__ void transpose_mat(const float* __restrict__ in, float* __restrict__ out,
                              int R, int Cc)
{
    const int i = blockIdx.x * blockDim.x + threadIdx.x;
    if (i >= R * Cc) return;
    const int r = i / Cc, c = i % Cc;
    out[c * R + r] = in[i];
}

// ---------------------------------------------------------------------------
// Mamba pieces
// ---------------------------------------------------------------------------
// u = xp[:, :ED] * sigmoid(xp[:, ED:2ED])
__global__ void glu_kernel(const float* __restrict__ xp, float* __restrict__ u, int total)
{
    const int i = blockIdx.x * blockDim.x + threadIdx.x;
    if (i >= total) return;
    const int c = i % EDC;
    const int row = i / EDC;
    const float a = xp[(size_t)row * (2 * EDC) + c];
    const float g = xp[(size_t)row * (2 * EDC) + EDC + c];
    u[i] = a * sigmoid_(g);
}

// depthwise causal conv (k=4, left pad 3) + SiLU; layout [B,T,ED]
__global__ void dwconv_silu(const float* __restrict__ u, const float* __restrict__ cw,
                            float* __restrict__ out, int total)
{
    const int i = blockIdx.x * blockDim.x + threadIdx.x;
    if (i >= total) return;
    const int c = i % EDC;
    const int t = (i / EDC) % TT;
    const int b = i / (EDC * TT);
    const float* base = u + (size_t)b * TT * EDC + c;
    float s = 0.f;
#pragma unroll
    for (int k = 0; k < DCC; ++k) {
        const int tt = t - (DCC - 1) + k;
        if (tt >= 0) s += base[(size_t)tt * EDC] * cw[c * DCC + k];
    }
    out[i] = s * sigmoid_(s);
}

// Selective-scan: one lane per (b, ed); 16-element state lives in VGPRs.
__global__ void mamba_scan(const float* __restrict__ u2,     // [B,T,ED]
                           const float* __restrict__ delta,  // [B,T,DS]
                           const float* __restrict__ Ap,     // [ED,DS]
                           const float* __restrict__ Bm,     // [ED,DS]
                           const float* __restrict__ Cm,     // [ED,DS]
                           const float* __restrict__ Dm,     // [ED]
                           float* __restrict__ y)            // [B,T,ED]
{
    const int b  = blockIdx.x;
    const int ed = threadIdx.x;
    float An[DSC], Bn[DSC], Cn[DSC], h[DSC];
#pragma unroll
    for (int n = 0; n < DSC; ++n) {
        An[n] = -__expf(Ap[ed * DSC + n]);
        Bn[n] = Bm[ed * DSC + n];
        Cn[n] = Cm[ed * DSC + n];
        h[n]  = 0.f;
    }
    const float D = Dm[ed];
    const float* ub = u2 + (size_t)b * TT * EDC + ed;
    const float* db = delta + (size_t)b * TT * DSC;
    float* yb = y + (size_t)b * TT * EDC + ed;
    for (int t = 0; t < TT; ++t) {
        const float uval = ub[(size_t)t * EDC];
        float acc = 0.f;
#pragma unroll
        for (int n = 0; n < DSC; ++n) {
            const float d = db[t * DSC + n];
            const float a = __expf(An[n] * d);
            h[n] = a * h[n] + d * Bn[n] * uval;
            acc += h[n] * Cn[n];
        }
        yb[(size_t)t * EDC] = acc + D * uval;
    }
}

// LayerNorm(gelu(x)) over last dim = 64; one row per thread, row in registers
__global__ void gelu_ln64(const float* __restrict__ in,
                          const float* __restrict__ g, const float* __restrict__ b,
                          float* __restrict__ out, int rows)
{
    const int r = blockIdx.x * blockDim.x + threadIdx.x;
    if (r >= rows) return;
    const float* x = in + (size_t)r * DIMC;
    float v[DIMC];
    float mu = 0.f;
#pragma unroll
    for (int i = 0; i < DIMC; ++i) {
        const float t = gelu_exact(x[i]);
        v[i] = t;
        mu += t;
    }
    mu *= (1.f / DIMC);
    float var = 0.f;
#pragma unroll
    for (int i = 0; i < DIMC; ++i) {
        const float d = v[i] - mu;
        var += d * d;
    }
    var *= (1.f / DIMC);
    const float inv = rsqrtf(var + 1e-5f);
#pragma unroll
    for (int i = 0; i < DIMC; ++i)
        out[(size_t)r * DIMC + i] = (v[i] - mu) * inv * g[i] + b[i];
}

// mean over T for [B,T,C] channel-last layout
__global__ void mean_time(const float* __restrict__ in, float* __restrict__ out, int C_)
{
    const int i = blockIdx.x * blockDim.x + threadIdx.x;
    if (i >= BB * C_) return;
    const int b = i / C_, c = i % C_;
    float s = 0.f;
    for (int t = 0; t < TT; ++t) s += in[((size_t)b * TT + t) * C_ + c];
    out[i] = s * (1.f / TT);
}

// mean over last (contiguous) dim
__global__ void mean_last(const float* __restrict__ in, float* __restrict__ out,
                          int rows, int L)
{
    const int i = blockIdx.x * blockDim.x + threadIdx.x;
    if (i >= rows) return;
    float s = 0.f;
    for (int t = 0; t < L; ++t) s += in[(size_t)i * L + t];
    out[i] = s / (float)L;
}

// ---------------------------------------------------------------------------
// Param-feature conv path
// ---------------------------------------------------------------------------
__global__ void conv1_bn_gelu(const float* __restrict__ x, const float* __restrict__ w,
                              const float* __restrict__ bias,
                              const float* __restrict__ g, const float* __restrict__ bb,
                              const float* __restrict__ m, const float* __restrict__ vv,
                              float* __restrict__ out)
{
    const int i = blockIdx.x * blockDim.x + threadIdx.x;   // B*32*T
    if (i >= BB * 32 * TT) return;
    const int t = i % TT;
    const int o = (i / TT) % 32;
    const int b = i / (TT * 32);
    float s = bias[o];
    for (int ic = 0; ic < PP; ++ic) {
#pragma unroll
        for (int k = 0; k < 3; ++k) {
            const int tt = t + k - 1;
            if (tt >= 0 && tt < TT)
                s += x[((size_t)b * TT + tt) * PP + ic] * w[(o * PP + ic) * 3 + k];
        }
    }
    const float sc = g[o] * rsqrtf(vv[o] + 1e-5f);
    s = (s - m[o]) * sc + bb[o];
    out[i] = gelu_exact(s);
}

__global__ void maxpool2(const float* __restrict__ in, float* __restrict__ out, int total)
{
    const int i = blockIdx.x * blockDim.x + threadIdx.x;
    if (i >= total) return;
    const int t = i % 256;
    const int row = i / 256;
    const float* p = in + (size_t)row * 512 + 2 * t;
    out[i] = fmaxf(p[0], p[1]);
}

__global__ void conv2_bn_gelu(const float* __restrict__ in, const float* __restrict__ w,
                              const float* __restrict__ bias,
                              const float* __restrict__ g, const float* __restrict__ bb,
                              const float* __restrict__ m, const float* __restrict__ vv,
                              float* __restrict__ out)
{
    const int i = blockIdx.x * blockDim.x + threadIdx.x;   // B*64*256
    if (i >= BB * 64 * 256) return;
    const int t = i % 256;
    const int o = (i / 256) % 64;
    const int b = i / (256 * 64);
    float s = bias[o];
    for (int ic = 0; ic < 32; ++ic) {
        const float* row = in + ((size_t)b * 32 + ic) * 256;
#pragma unroll
        for (int k = 0; k < 3; ++k) {
            const int tt = t + k - 1;
            if (tt >= 0 && tt < 256) s += row[tt] * w[(o * 32 + ic) * 3 + k];
        }
    }
    const float sc = g[o] * rsqrtf(vv[o] + 1e-5f);
    s = (s - m[o]) * sc + bb[o];
    out[i] = gelu_exact(s);
}

// ---------------------------------------------------------------------------
// Physics MLP 5->16->32, one lane per batch element
// ---------------------------------------------------------------------------
__global__ void physics_mlp(const float* __restrict__ x5,
                            const float* __restrict__ w1, const float* __restrict__ b1,
                            const float* __restrict__ w2, const float* __restrict__ b2,
                            float* __restrict__ ph)
{
    const int b = threadIdx.x;
    if (b >= BB) return;
    float h[16];
#pragma unroll
    for (int j = 0; j < 16; ++j) {
        float s = b1[j];
        for (int i = 0; i < 5; ++i) s += x5[b * 5 + i] * w1[j * 5 + i];
        h[j] = gelu_exact(s);
    }
    for (int o = 0; o < 32; ++o) {
        float s = b2[o];
#pragma unroll
        for (int j = 0; j < 16; ++j) s += h[j] * w2[o * 16 + j];
        ph[b * 32 + o] = gelu_exact(s);
    }
}

__global__ void concat_feats(const float* __restrict__ tf, const float* __restrict__ pf,
                             const float* __restrict__ ph, float* __restrict__ f0)
{
    const int i = blockIdx.x * blockDim.x + threadIdx.x;
    if (i >= BB * 224) return;
    const int b = i / 224, c = i % 224;
    float v;
    if (c < 128)      v = tf[b * 128 + c];
    else if (c < 192) v = pf[b * 64 + (c - 128)];
    else              v = ph[b * 32 + (c - 192)];
    f0[i] = v;
}

__global__ void copy_f_out(const float* __restrict__ f, float* __restrict__ out)
{
    const int i = blockIdx.x * blockDim.x + threadIdx.x;
    if (i >= BB * 128) return;
    out[BB * 12 + i] = f[i];     // logits (32*12) first, then f
}

// ---------------------------------------------------------------------------
// Host orchestration
// ---------------------------------------------------------------------------
struct MambaParams {
    const float *inp, *cw, *A, *Bm, *Cm, *Dm, *outp;
};

extern "C" void kernel_launch(void* const* d_in, const int* in_sizes, int n_in,
                              void* d_out, int out_size, void* d_ws, size_t ws_size,
                              hipStream_t stream)
{
    (void)in_sizes; (void)n_in; (void)out_size; (void)ws_size;
    int idx = 0;
    const float* x   = (const float*)d_in[idx++];   // [B,T,P]
    const float* phx = (const float*)d_in[idx++];   // [B,5]
    const float* lin1_w = (const float*)d_in[idx++];
    const float* lin1_b = (const float*)d_in[idx++];
    MambaParams m1, m2;
    m1.inp  = (const float*)d_in[idx++]; m1.cw  = (const float*)d_in[idx++];
    m1.A    = (const float*)d_in[idx++]; m1.Bm  = (const float*)d_in[idx++];
    m1.Cm   = (const float*)d_in[idx++]; m1.Dm  = (const float*)d_in[idx++];
    m1.outp = (const float*)d_in[idx++];
    const float* ln1_g = (const float*)d_in[idx++];
    const float* ln1_b = (const float*)d_in[idx++];
    m2.inp  = (const float*)d_in[idx++]; m2.cw  = (const float*)d_in[idx++];
    m2.A    = (const float*)d_in[idx++]; m2.Bm  = (const float*)d_in[idx++];
    m2.Cm   = (const float*)d_in[idx++]; m2.Dm  = (const float*)d_in[idx++];
    m2.outp = (const float*)d_in[idx++];
    const float* ln2_g = (const float*)d_in[idx++];
    const float* ln2_b = (const float*)d_in[idx++];
    const float* lin2_w = (const float*)d_in[idx++];
    const float* lin2_b = (const float*)d_in[idx++];
    const float* conv1_w = (const float*)d_in[idx++];
    const float* conv1_b = (const float*)d_in[idx++];
    const float* bn1_g = (const float*)d_in[idx++];
    const float* bn1_b = (const float*)d_in[idx++];
    const float* bn1_m = (const float*)d_in[idx++];
    const float* bn1_v = (const float*)d_in[idx++];
    const float* conv2_w = (const float*)d_in[idx++];
    const float* conv2_b = (const float*)d_in[idx++];
    const float* bn2_g = (const float*)d_in[idx++];
    const float* bn2_b = (const float*)d_in[idx++];
    const float* bn2_m = (const float*)d_in[idx++];
    const float* bn2_v = (const float*)d_in[idx++];
    const float* phys_w1 = (const float*)d_in[idx++];
    const float* phys_b1 = (const float*)d_in[idx++];
    const float* phys_w2 = (const float*)d_in[idx++];
    const float* phys_b2 = (const float*)d_in[idx++];
    const float* fus_w1 = (const float*)d_in[idx++];
    const float* fus_b1 = (const float*)d_in[idx++];
    const float* fbn1_g = (const float*)d_in[idx++];
    const float* fbn1_b = (const float*)d_in[idx++];
    const float* fbn1_m = (const float*)d_in[idx++];
    const float* fbn1_v = (const float*)d_in[idx++];
    const float* fus_w2 = (const float*)d_in[idx++];
    const float* fus_b2 = (const float*)d_in[idx++];
    const float* fbn2_g = (const float*)d_in[idx++];
    const float* fbn2_b = (const float*)d_in[idx++];
    const float* fbn2_m = (const float*)d_in[idx++];
    const float* fbn2_v = (const float*)d_in[idx++];
    const float* cls_w1 = (const float*)d_in[idx++];
    const float* cls_b1 = (const float*)d_in[idx++];
    const float* cls_w2 = (const float*)d_in[idx++];
    const float* cls_b2 = (const float*)d_in[idx++];

    // Workspace layout (floats); total ~12.9M floats (~49 MB)
    float* ws    = (float*)d_ws;
    float* T0    = ws;                  // [BT,64]
    float* XP    = ws + 1048576;        // [BT,256]; reused for conv scratch & t2
    float* U     = ws + 5242880;        // [BT,128]
    float* U2    = ws + 7340032;        // [BT,128]
    float* DELTA = ws + 9437184;        // [BT,16]
    float* Y     = ws + 9699328;        // [BT,128]
    float* TOUT  = ws + 11796480;       // [BT,64]
    float* SM    = ws + 12845056;
    float* TF  = SM;            // [32,128]
    float* PF  = SM + 4096;     // [32,64]
    float* PH  = SM + 6144;     // [32,32]
    float* F0  = SM + 7168;     // [32,224]
    float* F1  = SM + 14336;    // [32,256]
    float* FF  = SM + 22528;    // [32,128]
    float* G1  = SM + 26624;    // [32,64]
    float* BMT = SM + 28672;    // [16,128]  Bm transposed to [N,K]
    // conv scratch inside XP (conv path runs first)
    float* C1  = XP;            // [32,32,512]
    float* C1P = XP + 524288;   // [32,32,256]
    float* C2  = XP + 786432;   // [32,64,256]

    // ---- param_feature_extractor (uses XP region as scratch, runs first) ----
    {
        int n1 = BB * 32 * TT;
        conv1_bn_gelu<<<(n1 + 255) / 256, 256, 0, stream>>>(
            x, conv1_w, conv1_b, bn1_g, bn1_b, bn1_m, bn1_v, C1);
        int n2 = BB * 32 * 256;
        maxpool2<<<(n2 + 255) / 256, 256, 0, stream>>>(C1, C1P, n2);
        int n3 = BB * 64 * 256;
        conv2_bn_gelu<<<(n3 + 255) / 256, 256, 0, stream>>>(
            C1P, conv2_w, conv2_b, bn2_g, bn2_b, bn2_m, bn2_v, C2);
        mean_last<<<(BB * 64 + 255) / 256, 256, 0, stream>>>(C2, PF, BB * 64, 256);
    }

    // ---- physics_processor ----
    physics_mlp<<<1, 32, 0, stream>>>(phx, phys_w1, phys_b1, phys_w2, phys_b2, PH);

    // ---- time_feature_extractor ----
    gemm_launch<4, false>(x, PP, lin1_w, PP, lin1_b, nullptr, T0, DIMC,
                          BT, DIMC, PP, 0, nullptr, nullptr, nullptr, nullptr, stream);

    auto run_mamba = [&](const MambaParams& mp, const float* tin, float* tout) {
        // xp = t @ in_proj^T    [BT,256]
        gemm_launch<4, false>(tin, DIMC, mp.inp, DIMC, nullptr, nullptr, XP, 2 * EDC,
                              BT, 2 * EDC, DIMC, 0, nullptr, nullptr, nullptr, nullptr,
                              stream);
        const int ne = BT * EDC;
        glu_kernel<<<(ne + 255) / 256, 256, 0, stream>>>(XP, U, ne);
        dwconv_silu<<<(ne + 255) / 256, 256, 0, stream>>>(U, mp.cw, U2, ne);
        // delta = sigmoid(u2 @ Bm)   [BT,16]; Bm [128,16] -> BMT [16,128] (N,K)
        transpose_mat<<<(EDC * DSC + 255) / 256, 256, 0, stream>>>(mp.Bm, BMT, EDC, DSC);
        gemm_launch<1, false>(U2, EDC, BMT, EDC, nullptr, nullptr, DELTA, DSC,
                              BT, DSC, EDC, 2, nullptr, nullptr, nullptr, nullptr,
                              stream);
        mamba_scan<<<BB, EDC, 0, stream>>>(U2, DELTA, mp.A, mp.Bm, mp.Cm, mp.Dm, Y);
        // out = y @ out_proj^T + residual(t)
        gemm_launch<4, false>(Y, EDC, mp.outp, EDC, nullptr, tin, tout, DIMC,
                              BT, DIMC, EDC, 0, nullptr, nullptr, nullptr, nullptr,
                              stream);
    };

    run_mamba(m1, T0, TOUT);
    gelu_ln64<<<(BT + 127) / 128, 128, 0, stream>>>(TOUT, ln1_g, ln1_b, T0, BT);
    run_mamba(m2, T0, TOUT);
    gelu_ln64<<<(BT + 127) / 128, 128, 0, stream>>>(TOUT, ln2_g, ln2_b, T0, BT);

    // t2 = gelu(t @ lin2^T + b)  -> XP [BT,128]; then mean over T
    gemm_launch<4, false>(T0, DIMC, lin2_w, DIMC, lin2_b, nullptr, XP, 128,
                          BT, 128, DIMC, 1, nullptr, nullptr, nullptr, nullptr, stream);
    mean_time<<<(BB * 128 + 255) / 256, 256, 0, stream>>>(XP, TF, 128);

    // ---- fusion + classifier ----
    concat_feats<<<(BB * 224 + 255) / 256, 256, 0, stream>>>(TF, PF, PH, F0);
    gemm_launch<4, false>(F0, 224, fus_w1, 224, fus_b1, nullptr, F1, 256,
                          BB, 256, 224, 3, fbn1_g, fbn1_b, fbn1_m, fbn1_v, stream);
    gemm_launch<4, false>(F1, 256, fus_w2, 256, fus_b2, nullptr, FF, 128,
                          BB, 128, 256, 3, fbn2_g, fbn2_b, fbn2_m, fbn2_v, stream);
    gemm_launch<4, false>(FF, 128, cls_w1, 128, cls_b1, nullptr, G1, 64,
                          BB, 64, 128, 1, nullptr, nullptr, nullptr, nullptr, stream);
    gemm_launch<1, true>(G1, 64, cls_w2, 64, cls_b2, nullptr, (float*)d_out, 12,
                         BB, 12, 64, 0, nullptr, nullptr, nullptr, nullptr, stream);
    copy_f_out<<<(BB * 128 + 255) / 256, 256, 0, stream>>>(FF, (float*)d_out);
}